// LSSVoxelformer_87282325390014
// MI455X (gfx1250) — compile-verified
//
#include <hip/hip_runtime.h>

typedef float v2f __attribute__((ext_vector_type(2)));
typedef float v8f __attribute__((ext_vector_type(8)));

#define NQ     262144          // 16*128*128
#define CCH    128
#define IMGW   704.0f
#define IMGH   256.0f

// ---- workspace (float) layout ----
#define WS_S   0     // 128: bn scale (gamma*rsqrt(var+eps))
#define WS_T   128   // 128: bn shift (beta - mu*s)
#define WS_M   256   // 96 : world2img (6 cams x 4x4 row-major)
#define WS_G   352   // 9  : invd[3], vs[3], pcr_lo[3]

// ---- LDS (float) layout ----
#define L_S    0
#define L_T    128
#define L_M    256
#define L_G    352
#define L_W1   384                    // 3*128
#define L_B2   768                    // 128
#define L_WB   896                    // 128 rows * 130 (padded) = 16640
#define L_PROJ (L_WB + 16640)         // 8 waves * 96*3 = 2304
#define L_AGG  (L_PROJ + 8 * 288)     // 8 waves * 16*130 = 16640
#define L_TOTAL (L_AGG + 8 * 2080)    // = 36480 floats = 145920 bytes

// ---------------------------------------------------------------------------
// Prep: analytic BN statistics (grid is deterministic & separable -> mean/var
// closed form; b1 cancels inside BN), fused world2img matrices, geometry.
// ---------------------------------------------------------------------------
__global__ void lss_prep(const float* __restrict__ w2c, const float* __restrict__ c2i,
                         const float* __restrict__ vsz, const float* __restrict__ pcr,
                         const float* __restrict__ W1, const float* __restrict__ gamma,
                         const float* __restrict__ beta, float* __restrict__ ws)
{
    const int t = threadIdx.x;
    const float vs0 = vsz[0] * 2.0f, vs1 = vsz[1] * 2.0f, vs2 = vsz[2] * 2.0f;
    const float d0 = pcr[3] - pcr[0], d1 = pcr[4] - pcr[1], d2 = pcr[5] - pcr[2];
    const float inv0 = vs0 / d0, inv1 = vs1 / d1, inv2 = vs2 / d2;
    const float Nx = 128.0f, Ny = 128.0f, Nz = 16.0f;
    const float mean0 = inv0 * Nx * 0.5f, mean1 = inv1 * Ny * 0.5f, mean2 = inv2 * Nz * 0.5f;
    const float var0 = inv0 * inv0 * (Nx * Nx - 1.0f) * (1.0f / 12.0f);
    const float var1 = inv1 * inv1 * (Ny * Ny - 1.0f) * (1.0f / 12.0f);
    const float var2 = inv2 * inv2 * (Nz * Nz - 1.0f) * (1.0f / 12.0f);

    if (t < 128) {
        const float w0 = W1[t], w1 = W1[128 + t], w2 = W1[256 + t];
        const float mu  = mean0 * w0 + mean1 * w1 + mean2 * w2;
        const float var = var0 * w0 * w0 + var1 * w1 * w1 + var2 * w2 * w2;
        const float s   = gamma[t] * rsqrtf(var + 1e-5f);
        ws[WS_S + t] = s;
        ws[WS_T + t] = beta[t] - mu * s;
    }
    if (t < 96) {  // world2img[n] = cam2img[n] @ world2cam[n]
        const int n = t >> 4, rc = t & 15, i = rc >> 2, j = rc & 3;
        float acc = 0.0f;
        #pragma unroll
        for (int k = 0; k < 4; ++k)
            acc += c2i[(n * 4 + i) * 4 + k] * w2c[(n * 4 + k) * 4 + j];
        ws[WS_M + t] = acc;
    }
    if (t == 0) {
        ws[WS_G + 0] = inv0; ws[WS_G + 1] = inv1; ws[WS_G + 2] = inv2;
        ws[WS_G + 3] = vs0;  ws[WS_G + 4] = vs1;  ws[WS_G + 5] = vs2;
        ws[WS_G + 6] = pcr[0]; ws[WS_G + 7] = pcr[1]; ws[WS_G + 8] = pcr[2];
    }
}

// ---------------------------------------------------------------------------
// Main fused kernel: MLP-GEMM1 (WMMA f32) -> projection -> multilevel bilinear
// gather -> camera aggregation -> GEMM2 (WMMA f32, chained into GEMM1 acc).
// 256 threads = 8 waves, 16 queries per wave, 128 queries per block.
// ---------------------------------------------------------------------------
__global__ __launch_bounds__(256)
void lss_main(const float* __restrict__ f0, const float* __restrict__ f1,
              const float* __restrict__ f2, const float* __restrict__ f3,
              const float* __restrict__ ws, const float* __restrict__ W1,
              const float* __restrict__ b2, const float* __restrict__ W2,
              const float* __restrict__ Wout, float* __restrict__ out)
{
    extern __shared__ float sm[];
    const int tid  = threadIdx.x;
    const int wave = tid >> 5;
    const int lane = tid & 31;
    const int m    = lane & 15;   // row (A) / col-in-tile (B) index
    const int half = lane >> 4;   // selects K pair within a K=4 step

    // ---- stage constants + transposed W2 into LDS ----
    if (tid < 128) {
        sm[L_S  + tid] = ws[WS_S + tid];
        sm[L_T  + tid] = ws[WS_T + tid];
        sm[L_B2 + tid] = b2[tid];
        sm[L_W1 + tid]       = W1[tid];
        sm[L_W1 + 128 + tid] = W1[128 + tid];
        sm[L_W1 + 256 + tid] = W1[256 + tid];
    }
    if (tid < 96) sm[L_M + tid] = ws[WS_M + tid];
    if (tid < 9)  sm[L_G + tid] = ws[WS_G + tid];
    for (int idx = tid; idx < 16384; idx += 256) {
        const int k = idx >> 7, n = idx & 127;
        sm[L_WB + n * 130 + k] = W2[idx];          // W2T[n][k], padded stride
    }
    __syncthreads();

    const int qbase = blockIdx.x * 128 + wave * 16;
    const int qm    = qbase + m;
    const float xin = (float(qm & 127) + 0.5f)        * sm[L_G + 0];
    const float yin = (float((qm >> 7) & 127) + 0.5f) * sm[L_G + 1];
    const float zin = (float(qm >> 14) + 0.5f)        * sm[L_G + 2];

    v8f acc[8] = {};   // D = 16 queries x 128 channels per wave

    // ---- GEMM1: ve = relu(bn(ve_in @ W1)) @ W2  (A built on the fly) ----
    #pragma unroll 4
    for (int ks = 0; ks < 32; ++ks) {
        const int ka = ks * 4 + half * 2;
        const v2f w1x = *(const v2f*)&sm[L_W1 + ka];
        const v2f w1y = *(const v2f*)&sm[L_W1 + 128 + ka];
        const v2f w1z = *(const v2f*)&sm[L_W1 + 256 + ka];
        const v2f sc  = *(const v2f*)&sm[L_S + ka];
        const v2f tt  = *(const v2f*)&sm[L_T + ka];
        v2f a;
        a.x = fmaxf(sc.x * (xin * w1x.x + yin * w1y.x + zin * w1z.x) + tt.x, 0.0f);
        a.y = fmaxf(sc.y * (xin * w1x.y + yin * w1y.y + zin * w1z.y) + tt.y, 0.0f);
        #pragma unroll
        for (int ct = 0; ct < 8; ++ct) {
            const v2f bfrag = *(const v2f*)&sm[L_WB + (ct * 16 + m) * 130 + ka];
            acc[ct] = __builtin_amdgcn_wmma_f32_16x16x4_f32(
                false, a, false, bfrag, (short)0, acc[ct], false, false);
        }
    }
    __syncthreads();   // every wave done reading W2T

    // ---- projection: 16 queries x 6 cameras per wave ----
    float* proj = &sm[L_PROJ + wave * 288];
    {
        const float vsx = sm[L_G + 3], vsy = sm[L_G + 4], vsz = sm[L_G + 5];
        const float ox  = sm[L_G + 6], oy  = sm[L_G + 7], oz  = sm[L_G + 8];
        #pragma unroll
        for (int it = 0; it < 3; ++it) {
            const int item = it * 32 + lane;           // 0..95
            const int ql = item / 6, n = item - ql * 6;
            const int q  = qbase + ql;
            const float wx = (float(q & 127) + 0.5f)        * vsx + ox;
            const float wy = (float((q >> 7) & 127) + 0.5f) * vsy + oy;
            const float wz = (float(q >> 14) + 0.5f)        * vsz + oz;
            const float* M = &sm[L_M + n * 16];
            const float cx = M[0] * wx + M[1] * wy + M[2]  * wz + M[3];
            const float cy = M[4] * wx + M[5] * wy + M[6]  * wz + M[7];
            const float cd = M[8] * wx + M[9] * wy + M[10] * wz + M[11];
            const float dm = fmaxf(cd, 1e-5f);
            const float xn = cx / dm * (1.0f / IMGW);
            const float yn = cy / dm * (1.0f / IMGH);
            const bool ok = (cd > 1e-5f) && (xn > 0.0f) && (xn < 1.0f)
                                         && (yn > 0.0f) && (yn < 1.0f);
            proj[item * 3 + 0] = xn;
            proj[item * 3 + 1] = yn;
            proj[item * 3 + 2] = ok ? 1.0f : 0.0f;
        }
    }
    __syncthreads();
    if (lane < 16) {   // normalize camera weights, fold in 1/4 level mean
        float ssum = 0.0f;
        #pragma unroll
        for (int n = 0; n < 6; ++n) ssum += proj[(lane * 6 + n) * 3 + 2];
        const float wn = 0.25f / fmaxf(ssum, 1.0f);
        #pragma unroll
        for (int n = 0; n < 6; ++n) proj[(lane * 6 + n) * 3 + 2] *= wn;
    }
    __syncthreads();

    // warm W_out toward cache for the restage (global_prefetch_b8)
    __builtin_prefetch(&Wout[tid * 64], 0, 1);

    // ---- multi-level bilinear gather, mask-weighted camera aggregation ----
    float* aggw = &sm[L_AGG + wave * 2080];
    {
        const float* fptr[4] = { f0, f1, f2, f3 };
        for (int ql = 0; ql < 16; ++ql) {
            float a0 = 0.0f, a1 = 0.0f, a2 = 0.0f, a3 = 0.0f;
            for (int n = 0; n < 6; ++n) {
                const float w = proj[(ql * 6 + n) * 3 + 2];   // wave-uniform
                if (w > 0.0f) {
                    const float xn = proj[(ql * 6 + n) * 3 + 0];
                    const float yn = proj[(ql * 6 + n) * 3 + 1];
                    #pragma unroll
                    for (int lvl = 0; lvl < 4; ++lvl) {
                        const int Wl = 88 >> lvl, Hl = 32 >> lvl;
                        const float px = xn * float(Wl) - 0.5f;
                        const float py = yn * float(Hl) - 0.5f;
                        const float xf = floorf(px), yf = floorf(py);
                        const float fx = px - xf, fy = py - yf;
                        int x0 = (int)xf; x0 = min(max(x0, 0), Wl - 1);
                        const int x1 = min(x0 + 1, Wl - 1);
                        int y0 = (int)yf; y0 = min(max(y0, 0), Hl - 1);
                        const int y1 = min(y0 + 1, Hl - 1);
                        const float w00 = (1.0f - fx) * (1.0f - fy) * w;
                        const float w01 = fx * (1.0f - fy) * w;
                        const float w10 = (1.0f - fx) * fy * w;
                        const float w11 = fx * fy * w;
                        const float* f = fptr[lvl];
                        #pragma unroll
                        for (int cc = 0; cc < 4; ++cc) {
                            const int c  = lane + cc * 32;
                            const int r0 = ((n * CCH + c) * Hl + y0) * Wl;
                            const int r1 = ((n * CCH + c) * Hl + y1) * Wl;
                            const float v = w00 * f[r0 + x0] + w01 * f[r0 + x1]
                                          + w10 * f[r1 + x0] + w11 * f[r1 + x1];
                            if      (cc == 0) a0 += v;
                            else if (cc == 1) a1 += v;
                            else if (cc == 2) a2 += v;
                            else              a3 += v;
                        }
                    }
                }
            }
            aggw[ql * 130 + lane]      = a0;
            aggw[ql * 130 + lane + 32] = a1;
            aggw[ql * 130 + lane + 64] = a2;
            aggw[ql * 130 + lane + 96] = a3;
        }
    }

    // ---- restage W_out (transposed) into the same LDS buffer ----
    for (int idx = tid; idx < 16384; idx += 256) {
        const int k = idx >> 7, n = idx & 127;
        sm[L_WB + n * 130 + k] = Wout[idx];
    }
    __syncthreads();

    // ---- GEMM2: acc += agg @ W_out  (chained into GEMM1 accumulators) ----
    #pragma unroll 4
    for (int ks = 0; ks < 32; ++ks) {
        const int ka = ks * 4 + half * 2;
        const v2f a = *(const v2f*)&aggw[m * 130 + ka];
        #pragma unroll
        for (int ct = 0; ct < 8; ++ct) {
            const v2f bfrag = *(const v2f*)&sm[L_WB + (ct * 16 + m) * 130 + ka];
            acc[ct] = __builtin_amdgcn_wmma_f32_16x16x4_f32(
                false, a, false, bfrag, (short)0, acc[ct], false, false);
        }
    }

    // ---- epilogue: +b2, store channel-major [C, Z*Y*X] as 2x b128 per tile ----
    const int qrow = qbase + half * 8;
    #pragma unroll
    for (int ct = 0; ct < 8; ++ct) {
        const int c  = ct * 16 + m;
        const float bb = sm[L_B2 + c];
        float4 lo = make_float4(acc[ct][0] + bb, acc[ct][1] + bb,
                                acc[ct][2] + bb, acc[ct][3] + bb);
        float4 hi = make_float4(acc[ct][4] + bb, acc[ct][5] + bb,
                                acc[ct][6] + bb, acc[ct][7] + bb);
        float* dst = out + (size_t)c * NQ + qrow;
        *(float4*)(dst)     = lo;
        *(float4*)(dst + 4) = hi;
    }
}

extern "C" void kernel_launch(void* const* d_in, const int* in_sizes, int n_in,
                              void* d_out, int out_size, void* d_ws, size_t ws_size,
                              hipStream_t stream) {
    (void)in_sizes; (void)n_in; (void)out_size; (void)ws_size;
    const float* f0    = (const float*)d_in[0];
    const float* f1    = (const float*)d_in[1];
    const float* f2    = (const float*)d_in[2];
    const float* f3    = (const float*)d_in[3];
    const float* w2c   = (const float*)d_in[4];
    const float* c2i   = (const float*)d_in[5];
    // d_in[6] = img (unused by the reference forward)
    const float* vsz   = (const float*)d_in[7];
    const float* pcr   = (const float*)d_in[8];
    const float* W1    = (const float*)d_in[9];
    // d_in[10] = b1 (cancels inside BatchNorm)
    const float* gamma = (const float*)d_in[11];
    const float* beta  = (const float*)d_in[12];
    const float* W2    = (const float*)d_in[13];
    const float* b2    = (const float*)d_in[14];
    const float* Wout  = (const float*)d_in[15];
    float* ws  = (float*)d_ws;
    float* out = (float*)d_out;

    lss_prep<<<dim3(1), dim3(128), 0, stream>>>(w2c, c2i, vsz, pcr, W1, gamma, beta, ws);

    const size_t smem = (size_t)L_TOTAL * sizeof(float);   // ~143 KB < 320 KB/WGP
    lss_main<<<dim3(NQ / 128), dim3(256), smem, stream>>>(
        f0, f1, f2, f3, ws, W1, b2, W2, Wout, out);
}